// MeanPoolDownsample_21114059227744
// MI455X (gfx1250) — compile-verified
//
#include <hip/hip_runtime.h>

// Problem constants (match reference).
constexpr int kNumFine = 1048576;      // rows
constexpr int kFeat    = 128;          // features per row
constexpr int kSeg     = 262144;       // output segments
constexpr int kOutF4   = kSeg * (kFeat / 4);   // 8,388,608 float4 in output
constexpr int kCntF4   = kSeg / 4;             // 65,536 float4 in count buffer

typedef float v4f __attribute__((ext_vector_type(4)));

// ---------------------------------------------------------------------------
// Kernel 1: zero the segment-sum buffer (d_out) and the count buffer (d_ws).
// b128 stores; regular temporal hint so the zeroed lines are resident in L2
// for the atomic pass (128 MB accumulator < 192 MB L2).
// ---------------------------------------------------------------------------
__global__ __launch_bounds__(256) void zero_init(v4f* __restrict__ out,
                                                 v4f* __restrict__ cnt) {
    const int i = blockIdx.x * 256 + threadIdx.x;
    const v4f z = {0.f, 0.f, 0.f, 0.f};
    if (i < kOutF4) out[i] = z;
    if (i < kCntF4) cnt[i] = z;
}

// ---------------------------------------------------------------------------
// Kernel 2: scatter-add. One wave32 per input row: 32 lanes x float4 = 128
// features. Input is streamed once -> non-temporal loads keep the 512 MB
// stream from evicting the L2-resident accumulator. Segment id is
// wave-uniform -> readfirstlane puts it in an SGPR so the atomics use the
// saddr+voffset global form. Atomics are no-return -> STOREcnt path.
// ---------------------------------------------------------------------------
__global__ __launch_bounds__(256) void scatter_add(
    const v4f* __restrict__ in,        // [kNumFine * 32] float4
    const int* __restrict__ map,       // [kNumFine]
    float* __restrict__ out,           // [kSeg * kFeat] sums
    float* __restrict__ cnt) {         // [kSeg] counts
    const int wave = threadIdx.x >> 5;
    const int lane = threadIdx.x & 31;
    const int row  = blockIdx.x * 8 + wave;     // 8 waves per 256-thread block

    // Wave-uniform segment id -> SGPR.
    const int seg = __builtin_amdgcn_readfirstlane(map[row]);

    // 16B per lane, whole row per wave. NT: don't pollute L2.
    const v4f v = __builtin_nontemporal_load(&in[row * 32 + lane]);

    float* __restrict__ dst = out + (size_t)seg * kFeat + lane * 4;
    atomicAdd(dst + 0, v.x);
    atomicAdd(dst + 1, v.y);
    atomicAdd(dst + 2, v.z);
    atomicAdd(dst + 3, v.w);

    if (lane == 0) {
        atomicAdd(cnt + seg, 1.0f);
    }
}

// ---------------------------------------------------------------------------
// Kernel 3: mean = sum / max(count, 1). One thread per float4 of output.
// Sums are still L2-resident; final result stored non-temporal (write-once).
// ---------------------------------------------------------------------------
__global__ __launch_bounds__(256) void finalize_mean(
    v4f* __restrict__ out, const float* __restrict__ cnt) {
    const int i   = blockIdx.x * 256 + threadIdx.x;   // [0, kOutF4)
    const int seg = i >> 5;                           // 32 float4 per segment
    const float c   = cnt[seg];
    const float inv = 1.0f / fmaxf(c, 1.0f);
    v4f v = out[i];
    v.x *= inv; v.y *= inv; v.z *= inv; v.w *= inv;
    __builtin_nontemporal_store(v, &out[i]);
}

// ---------------------------------------------------------------------------
// Host-side launcher. d_in[0]=fine_scale_h_d (f32), d_in[1]=segment map (i32),
// d_out = [kSeg*kFeat] f32, d_ws >= kSeg*4 bytes for counts.
// ---------------------------------------------------------------------------
extern "C" void kernel_launch(void* const* d_in, const int* in_sizes, int n_in,
                              void* d_out, int out_size, void* d_ws, size_t ws_size,
                              hipStream_t stream) {
    const v4f*  in  = (const v4f*)d_in[0];
    const int*  map = (const int*)d_in[1];
    float*      out = (float*)d_out;
    float*      cnt = (float*)d_ws;     // kSeg floats = 1 MiB of scratch

    // 1) Zero sums + counts (kOutF4 threads dominate).
    zero_init<<<kOutF4 / 256, 256, 0, stream>>>((v4f*)out, (v4f*)cnt);

    // 2) Scatter-add: one wave per row, 8 rows per 256-thread block.
    scatter_add<<<kNumFine / 8, 256, 0, stream>>>(in, map, out, cnt);

    // 3) Divide by counts.
    finalize_mean<<<kOutF4 / 256, 256, 0, stream>>>((v4f*)out, cnt);
}